// LDEPool1d_7928509628951
// MI455X (gfx1250) — compile-verified
//
#include <hip/hip_runtime.h>
#include <math.h>

// Problem constants (match reference)
#define Bn   8
#define Fd   256
#define Tt   1024
#define Cc   64
#define TCH  32            // t-chunk per workgroup
#define XS   (Fd + 4)      // LDS stride for x chunk (pad to dodge bank conflicts)
#define MS   (Fd + 4)      // LDS stride for mu
#define SS   (Cc + 4)      // LDS stride for xmu / r
#define EPSV 1e-9f

typedef __attribute__((ext_vector_type(2))) float v2f;
typedef __attribute__((ext_vector_type(8))) float v8f;

// D = A(16x4) * B(4x16) + C, fp32 WMMA.
// A layout: lane l (l<16): m=l, VGPR0=K0,VGPR1=K1 ; (l>=16): m=l-16, VGPR0=K2,VGPR1=K3
// B layout: lane l (l<16): n=l, VGPR0=K0,VGPR1=K1 ; (l>=16): n=l-16, VGPR0=K2,VGPR1=K3
// C/D layout: VGPR v: lanes 0-15 -> (M=v, N=lane); lanes 16-31 -> (M=v+8, N=lane-16)
static __device__ __forceinline__ v8f wmma_f32(v2f a, v2f b, v8f c) {
    return __builtin_amdgcn_wmma_f32_16x16x4_f32(
        /*neg_a=*/false, a, /*neg_b=*/false, b,
        /*c_mod=*/(short)0, c, /*reuse_a=*/false, /*reuse_b=*/false);
}

__launch_bounds__(256)
__global__ void lde_main(const float* __restrict__ x,    // (B, F, T)
                         const float* __restrict__ mu,   // (C, F)
                         const float* __restrict__ prec, // (C)
                         float* __restrict__ Facc,       // (B, C, F) accum
                         float* __restrict__ Nacc)       // (B, C)    accum
{
    __shared__ float lds_x [TCH * XS];   // xt chunk, (t, f)
    __shared__ float lds_mu[Cc  * MS];   // mu, (c, f)
    __shared__ float lds_s [TCH * SS];   // xmu -> llk -> r, (t, c)
    __shared__ float lds_x2[TCH];
    __shared__ float lds_mu2[Cc];
    __shared__ float lds_p2[Cc];

    const int b    = blockIdx.x / (Tt / TCH);
    const int tchk = blockIdx.x % (Tt / TCH);
    const int t0   = tchk * TCH;
    const int tid  = threadIdx.x;
    const int lane = tid & 31;
    const int wv   = tid >> 5;           // 8 waves
    const int ml   = lane & 15;          // row-in-tile / col-in-tile lane index
    const int kh   = (lane >> 4) << 1;   // k sub-offset: 0 or 2
    const int mh   = (lane >> 4) << 3;   // C/D row offset: 0 or 8

    // ---- Stage x chunk: xt[t][f] = x[b][f][t0+t]; consecutive tids read
    // consecutive t (t is the contiguous axis of x) -> coalesced.
    const float* xb = x + (size_t)b * Fd * Tt;
    for (int i = tid; i < TCH * Fd; i += 256) {
        int f = i >> 5;          // i / TCH
        int t = i & (TCH - 1);   // i % TCH
        lds_x[t * XS + f] = xb[f * Tt + t0 + t];
    }
    // ---- Stage mu (L2-hot after first blocks)
    for (int i = tid; i < Cc * Fd; i += 256) {
        int c = i >> 8;          // i / Fd
        int f = i & (Fd - 1);    // i % Fd
        lds_mu[c * MS + f] = mu[i];
    }
    __syncthreads();

    // ---- per-row ||x||^2, per-comp ||mu||^2 and prec^2
    if (tid < TCH) {
        float s = 0.f;
        for (int f = 0; f < Fd; ++f) { float v = lds_x[tid * XS + f]; s += v * v; }
        lds_x2[tid] = s;
    } else if (tid >= 64 && tid < 64 + Cc) {
        int c = tid - 64;
        float s = 0.f;
        for (int f = 0; f < Fd; ++f) { float v = lds_mu[c * MS + f]; s += v * v; }
        lds_mu2[c] = s;
        float p = prec[c];
        lds_p2[c] = p * p;
    }

    // ================= GEMM1: xmu(t,c) = xt(t,f) * mu(c,f)^T =================
    // 32x64 output = 2 (t) x 4 (c) tiles of 16x16 -> one tile per wave.
    {
        const int m0 = (wv >> 2) * 16;   // t tile offset
        const int n0 = (wv & 3) * 16;    // c tile offset
        v8f acc = {};
        #pragma unroll 8
        for (int k0 = 0; k0 < Fd; k0 += 4) {
            v2f a, bb;
            const int k = k0 + kh;
            a.x  = lds_x [(m0 + ml) * XS + k + 0];
            a.y  = lds_x [(m0 + ml) * XS + k + 1];
            bb.x = lds_mu[(n0 + ml) * MS + k + 0];
            bb.y = lds_mu[(n0 + ml) * MS + k + 1];
            acc = wmma_f32(a, bb, acc);
        }
        __syncthreads();   // lds_x2/mu2/p2 done; also order vs lds_s writes
        #pragma unroll
        for (int v = 0; v < 8; ++v)
            lds_s[(m0 + v + mh) * SS + n0 + ml] = acc[v];
    }
    __syncthreads();

    // ================= llk + softmax over c =================
    for (int i = tid; i < TCH * Cc; i += 256) {
        int t = i >> 6, c = i & (Cc - 1);
        float d = lds_x2[t] - 2.f * lds_s[t * SS + c] + lds_mu2[c];
        lds_s[t * SS + c] = -lds_p2[c] * d;
    }
    __syncthreads();
    if (tid < TCH) {
        const int t = tid;
        float mx = -INFINITY;
        for (int c = 0; c < Cc; ++c) mx = fmaxf(mx, lds_s[t * SS + c]);
        float sum = 0.f;
        for (int c = 0; c < Cc; ++c) {
            float e = __expf(lds_s[t * SS + c] - mx);
            lds_s[t * SS + c] = e;
            sum += e;
        }
        float inv = 1.f / sum;
        for (int c = 0; c < Cc; ++c) lds_s[t * SS + c] *= inv;
    }
    __syncthreads();

    // ---- partial Nraw[c] = sum_t r[t][c]
    if (tid < Cc) {
        float s = 0.f;
        for (int t = 0; t < TCH; ++t) s += lds_s[t * SS + tid];
        atomicAdd(&Nacc[b * Cc + tid], s);
    }

    // ================= GEMM2: F(c,f) += r(t,c)^T * xt(t,f) =================
    // 64x256 output = 4 (c) x 16 (f) tiles -> 8 tiles per wave, K = TCH = 32.
    #pragma unroll
    for (int i = 0; i < 8; ++i) {
        const int tt = wv * 8 + i;
        const int m0 = (tt & 3) * 16;    // c tile offset
        const int n0 = (tt >> 2) * 16;   // f tile offset
        v8f acc = {};
        #pragma unroll
        for (int k0 = 0; k0 < TCH; k0 += 4) {
            v2f a, bb;
            const int k = k0 + kh;
            a.x  = lds_s[(k + 0) * SS + m0 + ml];   // A[m=c][k=t] = r[t][c]
            a.y  = lds_s[(k + 1) * SS + m0 + ml];
            bb.x = lds_x[(k + 0) * XS + n0 + ml];   // B[k=t][n=f] = xt[t][f]
            bb.y = lds_x[(k + 1) * XS + n0 + ml];
            acc = wmma_f32(a, bb, acc);
        }
        float* dst = &Facc[((size_t)b * Cc + m0 + mh) * Fd + n0 + ml];
        #pragma unroll
        for (int v = 0; v < 8; ++v)
            atomicAdd(dst + (size_t)v * Fd, acc[v]);
    }
}

__launch_bounds__(256)
__global__ void lde_finalize(const float* __restrict__ Facc,
                             const float* __restrict__ Nacc,
                             const float* __restrict__ mu,
                             float* __restrict__ out)
{
    int i = blockIdx.x * 256 + threadIdx.x;       // over B*C*F
    if (i >= Bn * Cc * Fd) return;
    int f = i & (Fd - 1);
    int c = (i >> 8) & (Cc - 1);
    int b = i >> 14;
    float Nr = Nacc[b * Cc + c];
    out[i] = (Facc[i] - Nr * mu[c * Fd + f]) / (Nr + EPSV);
}

extern "C" void kernel_launch(void* const* d_in, const int* in_sizes, int n_in,
                              void* d_out, int out_size, void* d_ws, size_t ws_size,
                              hipStream_t stream) {
    const float* x    = (const float*)d_in[0];   // (8, 256, 1024)
    const float* mu   = (const float*)d_in[1];   // (64, 256)
    const float* prec = (const float*)d_in[2];   // (64,)
    float* out  = (float*)d_out;                 // (8, 64*256)
    float* Facc = (float*)d_ws;                  // (8, 64, 256)
    float* Nacc = Facc + (size_t)Bn * Cc * Fd;   // (8, 64)

    hipMemsetAsync(d_ws, 0, ((size_t)Bn * Cc * Fd + Bn * Cc) * sizeof(float), stream);

    lde_main<<<Bn * (Tt / TCH), 256, 0, stream>>>(x, mu, prec, Facc, Nacc);
    lde_finalize<<<(Bn * Cc * Fd + 255) / 256, 256, 0, stream>>>(Facc, Nacc, mu, out);
}